// RankOrderPreservationLoss_27049704031074
// MI455X (gfx1250) — compile-verified
//
#include <hip/hip_runtime.h>
#include <math.h>

#define NN 320
#define D1 129            // 1 time + 128 spatial components
#define NTILE 20          // 320 / 16
#define MIN_DIST 0.1f
#define MARGIN 0.1f
#define WEIGHT 0.1f

typedef __attribute__((ext_vector_type(2))) float v2f;
typedef __attribute__((ext_vector_type(8))) float v8f;

// ---------------------------------------------------------------------------
// Kernel 1: Lorentz pairwise distances via V_WMMA_F32_16X16X4_F32.
// Spatial Gram S = sp . sp^T accumulated on the matrix pipe (128 = 32 clean
// k-steps of 4, zero masking / zero sign flips / full EXEC throughout), then
// the time component is applied in the epilogue:
//   x = t_i * t_j - S(i,j)   ;   dist = arccosh(max(x, 1+1e-7))
// Each wave computes a 16 x 64 strip: 4 output tiles sharing one A fragment.
// ---------------------------------------------------------------------------
__global__ __launch_bounds__(128)
void lorentz_dist_wmma(const float* __restrict__ emb, float* __restrict__ dist)
{
    const int lane = threadIdx.x & 31;
    const int wave = threadIdx.x >> 5;
    const int wid  = blockIdx.x * 4 + wave;   // 0..99
    const int tm   = wid / 5;                 // row tile 0..19
    const int tn0  = (wid % 5) * 4;           // first of 4 column tiles

    const int l16   = lane & 15;
    const int khalf = (lane >> 4) * 2;        // lanes 16-31 hold K+2,K+3

    // Per-lane spatial base pointers (skip the time component at k=0).
    const int mrow = tm * 16 + l16;
    const float* __restrict__ am = emb + mrow * D1 + 1 + khalf;
    const float* __restrict__ bm0 = emb + ((tn0 + 0) * 16 + l16) * D1 + 1 + khalf;
    const float* __restrict__ bm1 = emb + ((tn0 + 1) * 16 + l16) * D1 + 1 + khalf;
    const float* __restrict__ bm2 = emb + ((tn0 + 2) * 16 + l16) * D1 + 1 + khalf;
    const float* __restrict__ bm3 = emb + ((tn0 + 3) * 16 + l16) * D1 + 1 + khalf;

    v8f acc0 = {}, acc1 = {}, acc2 = {}, acc3 = {};

#pragma unroll 4
    for (int kb = 0; kb < 128; kb += 4) {
        v2f a, b0, b1, b2, b3;
        a[0]  = am[kb];  a[1]  = am[kb + 1];
        b0[0] = bm0[kb]; b0[1] = bm0[kb + 1];
        b1[0] = bm1[kb]; b1[1] = bm1[kb + 1];
        b2[0] = bm2[kb]; b2[1] = bm2[kb + 1];
        b3[0] = bm3[kb]; b3[1] = bm3[kb + 1];
        // (neg_a, A, neg_b, B, c_mod, C, reuse_a, reuse_b)
        acc0 = __builtin_amdgcn_wmma_f32_16x16x4_f32(false, a, false, b0,
                                                     (short)0, acc0, false, false);
        acc1 = __builtin_amdgcn_wmma_f32_16x16x4_f32(false, a, false, b1,
                                                     (short)0, acc1, false, false);
        acc2 = __builtin_amdgcn_wmma_f32_16x16x4_f32(false, a, false, b2,
                                                     (short)0, acc2, false, false);
        acc3 = __builtin_amdgcn_wmma_f32_16x16x4_f32(false, a, false, b3,
                                                     (short)0, acc3, false, false);
    }

    // Epilogue: time-component correction + arccosh, shared t_row across tiles.
    float trow[8];
#pragma unroll
    for (int r = 0; r < 8; ++r)
        trow[r] = emb[(tm * 16 + r + 8 * (lane >> 4)) * D1];  // t at k=0

#pragma unroll
    for (int t = 0; t < 4; ++t) {
        const int col = (tn0 + t) * 16 + l16;
        const float tcol = emb[col * D1];
        const v8f acc = (t == 0) ? acc0 : (t == 1) ? acc1 : (t == 2) ? acc2 : acc3;
#pragma unroll
        for (int r = 0; r < 8; ++r) {
            const int row = tm * 16 + r + 8 * (lane >> 4);
            float x = fmaxf(trow[r] * tcol - acc[r], 1.0f + 1e-7f);
            dist[row * NN + col] = logf(x + sqrtf(x * x - 1.0f));
        }
    }
}

// ---------------------------------------------------------------------------
// Kernel 2: per-anchor rank-order violation partial sums.
// Block i caches dist[i,:], gt[i,:], valid[i,:] in LDS, then strides all
// (j<k) pairs across 256 threads. Deterministic LDS tree reduction.
// ---------------------------------------------------------------------------
__global__ __launch_bounds__(256)
void rank_loss_partials(const float* __restrict__ dist,
                        const float* __restrict__ gt,
                        float* __restrict__ ptotal,
                        int*   __restrict__ pcount)
{
    __shared__ float s_emb[NN];
    __shared__ float s_gt[NN];
    __shared__ unsigned char s_valid[NN];
    __shared__ float s_rt[256];
    __shared__ int   s_rc[256];

    const int i   = blockIdx.x;
    const int tid = threadIdx.x;

    for (int j = tid; j < NN; j += 256) {
        const float g = gt[i * NN + j];
        s_emb[j]   = dist[i * NN + j];
        s_gt[j]    = g;
        s_valid[j] = (g >= MIN_DIST && j != i) ? 1 : 0;
    }
    __syncthreads();

    float tot = 0.0f;
    int   cnt = 0;
    for (int j = 0; j < NN - 1; ++j) {
        if (!s_valid[j]) continue;
        const float gj = s_gt[j];
        const float ej = s_emb[j];
        for (int k = j + 1 + tid; k < NN; k += 256) {
            if (!s_valid[k]) continue;
            const float gk = s_gt[k];
            const float s  = (gk > gj) ? 1.0f : ((gk < gj) ? -1.0f : 0.0f);
            if (s == 0.0f) continue;
            const float v = s * (ej - s_emb[k]) + MARGIN;  // relu'd below
            if (v > 0.0f) { tot += v; ++cnt; }
        }
    }

    s_rt[tid] = tot;
    s_rc[tid] = cnt;
    __syncthreads();
    for (int off = 128; off > 0; off >>= 1) {
        if (tid < off) { s_rt[tid] += s_rt[tid + off]; s_rc[tid] += s_rc[tid + off]; }
        __syncthreads();
    }
    if (tid == 0) { ptotal[i] = s_rt[0]; pcount[i] = s_rc[0]; }
}

// ---------------------------------------------------------------------------
// Kernel 3: fixed-order final reduction + epilogue.
// ---------------------------------------------------------------------------
__global__ __launch_bounds__(256)
void final_reduce(const float* __restrict__ ptotal,
                  const int*   __restrict__ pcount,
                  float* __restrict__ out)
{
    __shared__ float s_rt[256];
    __shared__ int   s_rc[256];
    const int tid = threadIdx.x;

    float tot = 0.0f;
    int   cnt = 0;
    for (int b = tid; b < NN; b += 256) { tot += ptotal[b]; cnt += pcount[b]; }
    s_rt[tid] = tot;
    s_rc[tid] = cnt;
    __syncthreads();
    for (int off = 128; off > 0; off >>= 1) {
        if (tid < off) { s_rt[tid] += s_rt[tid + off]; s_rc[tid] += s_rc[tid + off]; }
        __syncthreads();
    }
    if (tid == 0) {
        const float total = s_rt[0];
        const float count = (float)s_rc[0];
        out[0] = WEIGHT * ((count > 0.0f) ? total / fmaxf(count, 1.0f) : 0.0f);
    }
}

// ---------------------------------------------------------------------------
extern "C" void kernel_launch(void* const* d_in, const int* in_sizes, int n_in,
                              void* d_out, int out_size, void* d_ws, size_t ws_size,
                              hipStream_t stream)
{
    const float* embeddings = (const float*)d_in[0];   // 320 x 129
    const float* tree_dist  = (const float*)d_in[1];   // 320 x 320
    float* out = (float*)d_out;                        // scalar

    // Workspace layout: dist matrix | per-anchor totals | per-anchor counts
    float* dist   = (float*)d_ws;                      // NN*NN floats
    float* ptotal = dist + NN * NN;                    // NN floats
    int*   pcount = (int*)(ptotal + NN);               // NN ints

    // 100 waves (20 row tiles x 5 column strips of 4 tiles), 4 waves/block
    lorentz_dist_wmma<<<25, 128, 0, stream>>>(embeddings, dist);
    rank_loss_partials<<<NN, 256, 0, stream>>>(dist, tree_dist, ptotal, pcount);
    final_reduce<<<1, 256, 0, stream>>>(ptotal, pcount, out);
}